// HEDMForwardModel_35124242547426
// MI455X (gfx1250) — compile-verified
//
#include <hip/hip_runtime.h>
#include <hip/hip_bf16.h>
#include <math.h>

typedef float v2f __attribute__((ext_vector_type(2)));
typedef float v8f __attribute__((ext_vector_type(8)));

#define F_DEG2RAD 0.0174532925199433f
#define F_RAD2DEG 57.2957795130823f
#define F_EPS     1e-7f
#define F_PX      1.48f
#define F_OMEGA_STEP  0.25f
#define F_NFRAMES 1440.0f
#define F_NPIXY   2048.0f
#define F_NPIXZ   2048.0f
#define F_MINETA  (6.0f * F_DEG2RAD)
#define F_PI      3.14159265358979323846f
#define F_2PI     6.28318530717958647692f
#define F_INV2PI  0.15915494309189533577f
#define F_FRAME_SCALE (F_RAD2DEG / F_OMEGA_STEP)

// One wave32 computes a 16(n) x 16(m) tile.
// A-fragment (32-bit A 16x4, ISA 7.12.2): lanes 0-15: VGPR0=K0,VGPR1=K1 (row=lane);
//   lanes 16-31: VGPR0=K2, VGPR1=K3(pad 0) (row=lane-16).
// B-fragment (4x16): lanes 0-15: VGPR0=K0 row, VGPR1=K1 row (col=lane);
//   lanes 16-31: VGPR0=K2 row, VGPR1=K3(pad 0) (col=lane-16).
// C/D (16x16 f32): VGPR r: lanes 0-15 -> (n_local=r, m_local=lane);
//   lanes 16-31 -> (n_local=8+r, m_local=lane-16).
template<int DT>   // DT > 0: compile-time detector count; DT <= 0: runtime loop
__global__ __launch_bounds__(256) void hedm_fwd_kernel(
    const float* __restrict__ euler,   // N x 3
    const float* __restrict__ pos,     // N x 2
    const float* __restrict__ hkls,    // M x 3
    const float* __restrict__ thetas,  // M
    const float* __restrict__ Lsd,     // D
    const float* __restrict__ yBC,     // D
    const float* __restrict__ zBC,     // D
    float* __restrict__ out,
    int N, int M, int Drt, int tiles_m, int total_tiles)
{
    const int lane = threadIdx.x & 31;
    const int wave = threadIdx.x >> 5;
    const int tile = blockIdx.x * (blockDim.x >> 5) + wave;
    if (tile >= total_tiles) return;          // wave-uniform: EXEC stays all-ones

    const int tn = tile / tiles_m;
    const int tm = tile - tn * tiles_m;
    const int n0 = tn * 16;
    const int m0 = tm * 16;

    const int  nl = lane & 15;
    const bool hi = lane >= 16;

    // ---------- A fragments: per-lane rotation-matrix row values ----------
    int na = n0 + nl;
    float e0 = 0.f, e1 = 0.f, e2 = 0.f;
    if (na < N) { e0 = euler[na*3+0]; e1 = euler[na*3+1]; e2 = euler[na*3+2]; }
    float s0, c0, s1, c1, s2, c2;
    sincosf(e0, &s0, &c0);
    sincosf(e1, &s1, &c1);
    sincosf(e2, &s2, &c2);
    const float R00 =  c0*c2 - s0*c1*s2;
    const float R01 = -s0*c1*c2 - c0*s2;
    const float R02 =  s0*s1;
    const float R10 =  s0*c2 + c0*c1*s2;
    const float R11 =  c0*c1*c2 - s0*s2;
    const float R12 = -c0*s1;
    const float R20 =  s1*s2;
    const float R21 =  s1*c2;
    const float R22 =  c1;

    v2f a0, a1, a2;
    a0.x = hi ? R02 : R00;  a0.y = hi ? 0.0f : R01;
    a1.x = hi ? R12 : R10;  a1.y = hi ? 0.0f : R11;
    a2.x = hi ? R22 : R20;  a2.y = hi ? 0.0f : R21;

    // ---------- B fragment (shared across the three GEMMs) ----------
    int mb = m0 + nl;
    float h0 = 0.f, h1 = 0.f, h2 = 0.f;
    if (mb < M) { h0 = hkls[mb*3+0]; h1 = hkls[mb*3+1]; h2 = hkls[mb*3+2]; }
    v2f b;
    b.x = hi ? h2 : h0;  b.y = hi ? 0.0f : h1;

    // ---------- three f32 WMMAs: gx/gy/gz fragments in registers ----------
    v8f gx, gy, gz;
#pragma unroll
    for (int i = 0; i < 8; ++i) { gx[i] = 0.f; gy[i] = 0.f; gz[i] = 0.f; }
    gx = __builtin_amdgcn_wmma_f32_16x16x4_f32(false, a0, false, b, (short)0, gx, false, false);
    gy = __builtin_amdgcn_wmma_f32_16x16x4_f32(false, a1, false, b, (short)0, gy, false, false);
    gz = __builtin_amdgcn_wmma_f32_16x16x4_f32(false, a2, false, b, (short)0, gz, false, false);

    // ---------- hoisted per-lane / per-wave values ----------
    const int  m   = m0 + nl;
    const bool mok = (m < M);
    float st = 0.f, t2t = 0.f;
    if (mok) { float th = thetas[m]; st = sinf(th); t2t = tanf(2.0f * th); }

    constexpr int DMAX = (DT > 0) ? DT : 8;
    const int Dn = (DT > 0) ? DT : Drt;
    float Lds_r[DMAX], ybc_r[DMAX], zbc_r[DMAX];
#pragma unroll
    for (int d = 0; d < DMAX; ++d) {
        bool ok = (DT > 0) || (d < Drt);
        Lds_r[d] = ok ? Lsd[d] : 0.f;
        ybc_r[d] = ok ? yBC[d] : 0.f;
        zbc_r[d] = ok ? zBC[d] : 0.f;
    }

    const size_t S       = (size_t)N * 2u * (size_t)M;   // one (N,2,M) slab
    const size_t o_omega = 0;
    const size_t o_eta   = S;
    const size_t o_ypix  = 2*S;
    const size_t o_zpix  = 2*S + (size_t)Dn*S;
    const size_t o_frame = 2*S + 2*(size_t)Dn*S;
    const size_t o_valid = 3*S + 2*(size_t)Dn*S;

    const int nbase = n0 + (hi ? 8 : 0);

#pragma unroll
    for (int r = 0; r < 8; ++r) {
        const int n = nbase + r;
        const bool nok = (n < N);
        const float Gx = gx[r], Gy = gy[r], Gz = gz[r];

        const float rq     = Gx*Gx + Gy*Gy;
        const float glen   = sqrtf(rq + Gz*Gz);
        const float v      = st * glen;
        const float r0     = sqrtf(rq);
        const float cosarg = -v / (r0 + F_EPS);
        const bool  solok  = fabsf(cosarg) < 1.0f;
        const float ca     = fminf(fmaxf(cosarg, -1.0f + F_EPS), 1.0f - F_EPS);
        const float sd     = sqrtf(fmaxf(1.0f - ca*ca, 0.0f));   // sin(delta), delta=acos(ca)
        const float delta  = acosf(ca);
        const float phi    = atan2f(Gy, Gx);
        const float inv_r0 = 1.0f / fmaxf(r0, 1e-30f);

        // angle-addition terms: omega = -phi +/- delta
        const float Aт = Gx*ca*inv_r0, Bт = Gy*sd*inv_r0;
        const float Cт = Gy*ca*inv_r0, Dт = Gx*sd*inv_r0;

        float px = 0.f, py = 0.f;
        if (nok) { px = pos[n*2+0]; py = pos[n*2+1]; }

#pragma unroll
        for (int s = 0; s < 2; ++s) {
            float w = (s == 0) ? (-phi + delta) : (-phi - delta);
            // Python-style (w + pi) % (2*pi) - pi without fmodf slow path:
            float t = (w + F_PI) * F_INV2PI;
            t -= floorf(t);                    // in [0,1): divisor-sign modulo
            w = t * F_2PI - F_PI;

            // cos/sin of omega algebraically (wrap-invariant), no sincos:
            const float co = (s == 0) ? (Aт + Bт) : (Aт - Bт);
            const float so = (s == 0) ? (Dт - Cт) : (-Dт - Cт);

            const float gyp  = so*Gx + co*Gy;
            const float gzp  = Gz;
            const float elen = sqrtf(gyp*gyp + gzp*gzp) + F_EPS;
            const float ce   = fminf(fmaxf(gzp/elen, -1.0f + F_EPS), 1.0f - F_EPS);
            const float eta0 = acosf(ce);
            const bool  neg  = (gyp > 0.0f);
            const float eta  = neg ? -eta0 : eta0;
            const float se_m = sqrtf(fmaxf(1.0f - ce*ce, 0.0f)); // sin(|eta|)
            const float se   = neg ? -se_m : se_m;

            const float frame = w * F_FRAME_SCALE;
            const float xl = px*co - py*so;
            const float yl = px*so + py*co;
            const float tse = t2t * se;
            const float tce = t2t * ce;

            const size_t idx = ((size_t)n*2u + (size_t)s) * (size_t)M + (size_t)m;
            const bool store = mok && nok;

            bool all_layers = true;
#pragma unroll
            for (int d = 0; d < DMAX; ++d) {
                if (DT <= 0 && d >= Drt) break;
                const float Ld   = Lds_r[d] - xl;
                const float ypix = ybc_r[d] - (yl - Ld * tse) * (1.0f / F_PX); // FLIP_Y
                const float zpix = zbc_r[d] + (Ld * tce) * (1.0f / F_PX);
                const bool lv = (ypix >= 0.0f) && (ypix < F_NPIXY) &&
                                (zpix >= 0.0f) && (zpix < F_NPIXZ);
                all_layers = all_layers && lv;
                if (store) {
                    out[o_ypix + (size_t)d*S + idx] = ypix;
                    out[o_zpix + (size_t)d*S + idx] = zpix;
                }
            }

            const bool valid = solok && (fabsf(eta) > F_MINETA) &&
                               (frame >= 0.0f) && (frame < F_NFRAMES) && all_layers;
            if (store) {
                out[o_omega + idx] = w;
                out[o_eta   + idx] = eta;
                out[o_frame + idx] = frame;
                out[o_valid + idx] = valid ? 1.0f : 0.0f;
            }
        }
    }
}

extern "C" void kernel_launch(void* const* d_in, const int* in_sizes, int n_in,
                              void* d_out, int out_size, void* d_ws, size_t ws_size,
                              hipStream_t stream) {
    const float* euler  = (const float*)d_in[0];
    const float* pos    = (const float*)d_in[1];
    const float* hkls   = (const float*)d_in[2];
    const float* thetas = (const float*)d_in[3];
    const float* Lsd    = (const float*)d_in[4];
    const float* yBC    = (const float*)d_in[5];
    const float* zBC    = (const float*)d_in[6];
    float* out = (float*)d_out;

    const int N = in_sizes[0] / 3;
    const int M = in_sizes[3];
    const int D = in_sizes[4];

    const int tiles_n = (N + 15) / 16;
    const int tiles_m = (M + 15) / 16;
    const int total   = tiles_n * tiles_m;

    const int waves_per_block = 8;                    // 256 threads = 8 wave32
    const int blocks = (total + waves_per_block - 1) / waves_per_block;
    const dim3 grid(blocks), block(waves_per_block * 32);

    switch (D) {
    case 1:
        hedm_fwd_kernel<1><<<grid, block, 0, stream>>>(euler, pos, hkls, thetas, Lsd, yBC, zBC, out, N, M, D, tiles_m, total); break;
    case 2:
        hedm_fwd_kernel<2><<<grid, block, 0, stream>>>(euler, pos, hkls, thetas, Lsd, yBC, zBC, out, N, M, D, tiles_m, total); break;
    case 3:
        hedm_fwd_kernel<3><<<grid, block, 0, stream>>>(euler, pos, hkls, thetas, Lsd, yBC, zBC, out, N, M, D, tiles_m, total); break;
    case 4:
        hedm_fwd_kernel<4><<<grid, block, 0, stream>>>(euler, pos, hkls, thetas, Lsd, yBC, zBC, out, N, M, D, tiles_m, total); break;
    default:
        hedm_fwd_kernel<-1><<<grid, block, 0, stream>>>(euler, pos, hkls, thetas, Lsd, yBC, zBC, out, N, M, D, tiles_m, total); break;
    }
}